// GCNNModel_61203283968721
// MI455X (gfx1250) — compile-verified
//
#include <hip/hip_runtime.h>
#include <hip/hip_bf16.h>

typedef __attribute__((ext_vector_type(2))) float v2f;
typedef __attribute__((ext_vector_type(8))) float v8f;

// ---------------------------------------------------------------------------
// Degree / normalization kernels
// ---------------------------------------------------------------------------
__global__ void __launch_bounds__(256) deg_init_kernel(float* __restrict__ deg, int n) {
    int i = blockIdx.x * blockDim.x + threadIdx.x;
    if (i < n) deg[i] = 1.0f;   // self-loop contributes 1 to every node's degree
}

__global__ void __launch_bounds__(256) deg_count_kernel(float* __restrict__ deg,
                                                        const long long* __restrict__ dst,
                                                        int nedges) {
    int i = blockIdx.x * blockDim.x + threadIdx.x;
    if (i < nedges) {
        unsafeAtomicAdd(&deg[(size_t)dst[i]], 1.0f);
    }
}

__global__ void __launch_bounds__(256) deg_rsqrt_kernel(float* __restrict__ deg, int n) {
    int i = blockIdx.x * blockDim.x + threadIdx.x;
    if (i < n) deg[i] = rsqrtf(deg[i]);   // deg >= 1 always (self loops)
}

// ---------------------------------------------------------------------------
// GEMM: H = X @ W using V_WMMA_F32_16X16X4_F32 (exact fp32 math).
// One wave computes one 16x16 output tile; K = 64 in 16 steps of 4.
// Epilogue also writes AGG = H * dinv^2 (the self-loop message), so no
// separate agg-init pass is needed.
//
// Per-lane layouts (ISA 7.12.2, 32-bit operands):
//   A (16x4, MxK):  m = lane&15, kgrp = lane>>4; vgpr0 = A[m][2*kgrp],
//                   vgpr1 = A[m][2*kgrp+1]
//   B (4x16, KxN):  n = lane&15; vgpr0 = B[2*kgrp][n], vgpr1 = B[2*kgrp+1][n]
//   C/D (16x16):    vgpr r: lanes 0-15 -> C[r][lane], lanes 16-31 -> C[r+8][lane-16]
// ---------------------------------------------------------------------------
__global__ void __launch_bounds__(256) gcn_gemm_wmma_kernel(
        const float* __restrict__ X,    // [N,64] row-major
        const float* __restrict__ W,    // [64,64] row-major
        float* __restrict__ H,          // [N,64]
        float* __restrict__ AGG,        // [N,64]  = H * dinv^2
        const float* __restrict__ DINV, // [N]
        int nrows) {
    const int lane  = threadIdx.x & 31;
    const int wave  = threadIdx.x >> 5;           // 0..7
    const int tile  = blockIdx.x * 8 + wave;      // global tile index
    const int mtile = tile >> 2;                  // 4 N-tiles (64/16)
    const int ntile = tile & 3;
    const int row0  = mtile * 16;
    if (row0 >= nrows) return;                    // wave-uniform: EXEC stays all-1s

    const int m   = lane & 15;                    // M index (A) / N index (B,C)
    const int grp = lane >> 4;                    // 0 or 1 -> K sub-pair
    const int col0 = ntile * 16;

    const float* arow = X + (size_t)(row0 + m) * 64;

    v8f c = {};
#pragma unroll
    for (int k0 = 0; k0 < 64; k0 += 4) {
        const int ka = k0 + 2 * grp;
        v2f a, b;
        a.x = arow[ka];
        a.y = arow[ka + 1];
        b.x = W[(size_t)ka       * 64 + col0 + m];
        b.y = W[(size_t)(ka + 1) * 64 + col0 + m];
        // 8 args: (neg_a, A, neg_b, B, c_mod, C, reuse_a, reuse_b)
        c = __builtin_amdgcn_wmma_f32_16x16x4_f32(
                false, a, false, b, (short)0, c, false, false);
    }

#pragma unroll
    for (int r = 0; r < 8; ++r) {
        const int row = row0 + r + 8 * grp;
        const int col = col0 + m;
        const float v  = c[r];
        const float di = DINV[row];
        H[(size_t)row * 64 + col]   = v;
        AGG[(size_t)row * 64 + col] = v * di * di;
    }
}

// ---------------------------------------------------------------------------
// Edge scatter: AGG[dst] += H[src] * dinv[src]*dinv[dst]
// 16 threads per edge, float4 per thread -> fully coalesced 256B row access.
// ---------------------------------------------------------------------------
__global__ void __launch_bounds__(256) gcn_scatter_kernel(
        const float* __restrict__ H,
        float* __restrict__ AGG,
        const long long* __restrict__ src,
        const long long* __restrict__ dst,
        const float* __restrict__ DINV,
        int nedges) {
    int tid = blockIdx.x * blockDim.x + threadIdx.x;
    int e = tid >> 4;
    if (e >= nedges) return;
    int j = (tid & 15) * 4;

    const long long s = src[e];
    const long long d = dst[e];
    const float norm = DINV[(size_t)s] * DINV[(size_t)d];

    const float4 hv = *reinterpret_cast<const float4*>(H + (size_t)s * 64 + j);
    float* out = AGG + (size_t)d * 64 + j;
    unsafeAtomicAdd(out + 0, hv.x * norm);
    unsafeAtomicAdd(out + 1, hv.y * norm);
    unsafeAtomicAdd(out + 2, hv.z * norm);
    unsafeAtomicAdd(out + 3, hv.w * norm);
}

// ---------------------------------------------------------------------------
// OUT = relu(AGG + bias)
// ---------------------------------------------------------------------------
__global__ void __launch_bounds__(256) gcn_bias_relu_kernel(
        const float* __restrict__ AGG,
        const float* __restrict__ bias,
        float* __restrict__ OUT,
        int n) {
    int tid = blockIdx.x * blockDim.x + threadIdx.x;
    int i = tid >> 4;
    if (i >= n) return;
    int j = (tid & 15) * 4;

    float4 v = *reinterpret_cast<const float4*>(AGG + (size_t)i * 64 + j);
    float4 b = *reinterpret_cast<const float4*>(bias + j);
    v.x = fmaxf(v.x + b.x, 0.0f);
    v.y = fmaxf(v.y + b.y, 0.0f);
    v.z = fmaxf(v.z + b.z, 0.0f);
    v.w = fmaxf(v.w + b.w, 0.0f);
    *reinterpret_cast<float4*>(OUT + (size_t)i * 64 + j) = v;
}

// ---------------------------------------------------------------------------
// Launcher
// ---------------------------------------------------------------------------
extern "C" void kernel_launch(void* const* d_in, const int* in_sizes, int n_in,
                              void* d_out, int out_size, void* d_ws, size_t ws_size,
                              hipStream_t stream) {
    const float*     x  = (const float*)d_in[0];
    const long long* ei = (const long long*)d_in[1];   // int64 [2, E]
    const float*     W0 = (const float*)d_in[2];
    const float*     b0 = (const float*)d_in[3];
    const float*     W1 = (const float*)d_in[4];
    const float*     b1 = (const float*)d_in[5];
    const float*     W2 = (const float*)d_in[6];
    const float*     b2 = (const float*)d_in[7];

    const int N = in_sizes[0] / 64;   // 100000
    const int E = in_sizes[1] / 2;    // 1000000
    float* out = (float*)d_out;

    const long long* src = ei;
    const long long* dst = ei + E;

    // Workspace layout (floats): dinv[N] (padded) | H[N*64] | AGG[N*64] | TMP[N*64]
    float* ws   = (float*)d_ws;
    size_t npad = ((size_t)N + 255) & ~(size_t)255;
    float* dinv = ws;
    float* h    = ws + npad;
    float* agg  = h + (size_t)N * 64;
    float* tmp  = agg + (size_t)N * 64;

    const int B = 256;

    // ---- normalization: dinv = deg^{-1/2} with self loops ----
    deg_init_kernel<<<(N + B - 1) / B, B, 0, stream>>>(dinv, N);
    deg_count_kernel<<<(E + B - 1) / B, B, 0, stream>>>(dinv, dst, E);
    deg_rsqrt_kernel<<<(N + B - 1) / B, B, 0, stream>>>(dinv, N);

    const float* Ws[3]   = {W0, W1, W2};
    const float* bs[3]   = {b0, b1, b2};
    float*       outs[3] = {tmp, tmp, out};

    const int mtiles      = (N + 15) / 16;
    const int gemm_blocks = (mtiles * 4 + 7) / 8;           // 8 waves / block
    const int scat_blocks = (int)(((long long)E * 16 + B - 1) / B);
    const int relu_blocks = (int)(((long long)N * 16 + B - 1) / B);

    const float* in = x;
    for (int l = 0; l < 3; ++l) {
        gcn_gemm_wmma_kernel<<<gemm_blocks, B, 0, stream>>>(in, Ws[l], h, agg, dinv, N);
        gcn_scatter_kernel<<<scat_blocks, B, 0, stream>>>(h, agg, src, dst, dinv, E);
        gcn_bias_relu_kernel<<<relu_blocks, B, 0, stream>>>(agg, bs[l], outs[l], N);
        in = outs[l];
    }
}